// GM_GCN_13572096655876
// MI455X (gfx1250) — compile-verified
//
#include <hip/hip_runtime.h>

typedef __attribute__((ext_vector_type(2))) float v2f;
typedef __attribute__((ext_vector_type(8))) float v8f;

#define N_NODES 100000
#define N_EDGES 1600000
#define DIM 128
#define N_CLASSES 40
#define EPB 64  // edges staged per scatter block

// ---------- degree / normalization ----------
__global__ void deg_init_kernel(float* deg) {
    int i = blockIdx.x * blockDim.x + threadIdx.x;
    if (i < N_NODES) deg[i] = 1.0f;  // self loop contributes 1
}

__global__ void deg_acc_kernel(const int* __restrict__ col, float* deg) {
    int e = blockIdx.x * blockDim.x + threadIdx.x;
    if (e < N_EDGES) atomicAdd(&deg[col[e]], 1.0f);
}

__global__ void dinv_kernel(float* deg) {
    int i = blockIdx.x * blockDim.x + threadIdx.x;
    if (i < N_NODES) deg[i] = rsqrtf(deg[i]);  // deg >= 1 always
}

// ---------- WMMA GEMM: H[M x 128] = X[M x 128] @ W[128 x 128] ----------
// grid.x = M/16 tiles; block = 256 threads = 8 waves; wave w -> N-tile w.
__global__ __launch_bounds__(256) void gemm128_wmma(const float* __restrict__ X,
                                                    const float* __restrict__ W,
                                                    float* __restrict__ H) {
    const int m0    = blockIdx.x * 16;
    const int lane  = threadIdx.x & 31;
    const int wave  = threadIdx.x >> 5;
    const int n0    = wave * 16;
    const int l15   = lane & 15;
    const int khalf = (lane >> 4) << 1;  // 0 for lanes 0-15, 2 for lanes 16-31
    const float* xrow = X + (size_t)(m0 + l15) * DIM;

    v8f acc = {};
    for (int k = 0; k < DIM; k += 4) {
        const int kk = k + khalf;
        v2f a, b;
        a.x = xrow[kk];
        a.y = xrow[kk + 1];
        b.x = W[(size_t)kk * DIM + n0 + l15];
        b.y = W[(size_t)(kk + 1) * DIM + n0 + l15];
        acc = __builtin_amdgcn_wmma_f32_16x16x4_f32(
            /*neg_a=*/false, a, /*neg_b=*/false, b,
            /*c_mod=*/(short)0, acc, /*reuse_a=*/false, /*reuse_b=*/false);
    }
    const int crow = m0 + (lane >> 4) * 8;
    const int ccol = n0 + l15;
#pragma unroll
    for (int r = 0; r < 8; ++r)
        H[(size_t)(crow + r) * DIM + ccol] = acc[r];
}

// ---------- WMMA GEMM: OUT[M x 40] = X[M x 128] @ W[128 x 40] + bias ----------
// block = 96 threads = 3 waves covering N = 0..47 (cols >= 40 masked).
__global__ __launch_bounds__(96) void gemm_out_wmma(const float* __restrict__ X,
                                                    const float* __restrict__ W,
                                                    const float* __restrict__ bias,
                                                    float* __restrict__ OUT) {
    const int m0    = blockIdx.x * 16;
    const int lane  = threadIdx.x & 31;
    const int wave  = threadIdx.x >> 5;
    const int n0    = wave * 16;
    const int l15   = lane & 15;
    const int khalf = (lane >> 4) << 1;
    const int ncol  = n0 + l15;
    const bool nv   = (ncol < N_CLASSES);
    const float* xrow = X + (size_t)(m0 + l15) * DIM;

    v8f acc = {};
    for (int k = 0; k < DIM; k += 4) {
        const int kk = k + khalf;
        v2f a, b;
        a.x = xrow[kk];
        a.y = xrow[kk + 1];
        b.x = nv ? W[(size_t)kk * N_CLASSES + ncol] : 0.0f;
        b.y = nv ? W[(size_t)(kk + 1) * N_CLASSES + ncol] : 0.0f;
        acc = __builtin_amdgcn_wmma_f32_16x16x4_f32(
            false, a, false, b, (short)0, acc, false, false);
    }
    if (nv) {
        const int crow = m0 + (lane >> 4) * 8;
        const float bb = bias[ncol];
#pragma unroll
        for (int r = 0; r < 8; ++r)
            OUT[(size_t)(crow + r) * N_CLASSES + ncol] = acc[r] + bb;
    }
}

// ---------- aggregation ----------
// agg[i,:] = dinv[i]^2 * h[i,:]   (self-loop term; also zero-inits agg)
__global__ void agg_selfloop_kernel(const float* __restrict__ dinv,
                                    const float* __restrict__ h,
                                    float* __restrict__ agg) {
    size_t idx = (size_t)blockIdx.x * blockDim.x + threadIdx.x;
    if (idx < (size_t)N_NODES * DIM) {
        int node = (int)(idx >> 7);
        float d = dinv[node];
        agg[idx] = d * d * h[idx];
    }
}

// edge scatter: agg[col,:] += dinv[row]*dinv[col] * h[row,:]
// block = 128 threads (one feature per thread); EPB edges staged in LDS.
__global__ __launch_bounds__(128) void scatter_kernel(const int* __restrict__ row,
                                                      const int* __restrict__ col,
                                                      const float* __restrict__ dinv,
                                                      const float* __restrict__ h,
                                                      float* __restrict__ agg) {
    __shared__ int   s_row[EPB];
    __shared__ int   s_col[EPB];
    __shared__ float s_norm[EPB];
    const int e0 = blockIdx.x * EPB;
    const int t  = threadIdx.x;

    for (int i = t; i < EPB; i += 128) {
        int e = e0 + i;
        if (e < N_EDGES) {
            int r = row[e], c = col[e];
            s_row[i]  = r;
            s_col[i]  = c;
            s_norm[i] = dinv[r] * dinv[c];
        }
    }
    __syncthreads();

    int lim = N_EDGES - e0;
    if (lim > EPB) lim = EPB;
    for (int i = 0; i < lim; ++i) {
        float v = s_norm[i] * h[(size_t)s_row[i] * DIM + t];
        atomicAdd(&agg[(size_t)s_col[i] * DIM + t], v);
    }
}

// x = relu(agg + b)
__global__ void bias_relu_kernel(float* __restrict__ x, const float* __restrict__ b) {
    size_t idx = (size_t)blockIdx.x * blockDim.x + threadIdx.x;
    if (idx < (size_t)N_NODES * DIM)
        x[idx] = fmaxf(x[idx] + b[idx & (DIM - 1)], 0.0f);
}

extern "C" void kernel_launch(void* const* d_in, const int* in_sizes, int n_in,
                              void* d_out, int out_size, void* d_ws, size_t ws_size,
                              hipStream_t stream) {
    const float* x    = (const float*)d_in[0];
    const int*   edge = (const int*)d_in[1];
    const float* W1   = (const float*)d_in[2];
    const float* b1   = (const float*)d_in[3];
    const float* W2   = (const float*)d_in[4];
    const float* b2   = (const float*)d_in[5];
    const float* Wout = (const float*)d_in[6];
    const float* bout = (const float*)d_in[7];
    const int* row = edge;            // edge_index[0] = source j
    const int* col = edge + N_EDGES;  // edge_index[1] = target i
    float* out = (float*)d_out;

    char* ws = (char*)d_ws;
    float* dinv = (float*)ws;                              // N_NODES floats
    float* bufA = (float*)(ws + ((size_t)1 << 20));        // N_NODES*DIM (h)
    float* bufB = bufA + (size_t)N_NODES * DIM;            // N_NODES*DIM (agg/x)

    const int nelem  = N_NODES * DIM;
    const int ntiles = N_NODES / 16;  // 6250, exact

    // normalization
    deg_init_kernel<<<(N_NODES + 255) / 256, 256, 0, stream>>>(dinv);
    deg_acc_kernel<<<(N_EDGES + 255) / 256, 256, 0, stream>>>(col, dinv);
    dinv_kernel<<<(N_NODES + 255) / 256, 256, 0, stream>>>(dinv);

    // layer 1: h1 = x@W1 ; agg ; x1 = relu(agg + b1)
    gemm128_wmma<<<ntiles, 256, 0, stream>>>(x, W1, bufA);
    agg_selfloop_kernel<<<(nelem + 255) / 256, 256, 0, stream>>>(dinv, bufA, bufB);
    scatter_kernel<<<(N_EDGES + EPB - 1) / EPB, 128, 0, stream>>>(row, col, dinv, bufA, bufB);
    bias_relu_kernel<<<(nelem + 255) / 256, 256, 0, stream>>>(bufB, b1);

    // layer 2: h2 = x1@W2 ; agg ; x2 = relu(agg + b2)
    gemm128_wmma<<<ntiles, 256, 0, stream>>>(bufB, W2, bufA);
    agg_selfloop_kernel<<<(nelem + 255) / 256, 256, 0, stream>>>(dinv, bufA, bufB);
    scatter_kernel<<<(N_EDGES + EPB - 1) / EPB, 128, 0, stream>>>(row, col, dinv, bufA, bufB);
    bias_relu_kernel<<<(nelem + 255) / 256, 256, 0, stream>>>(bufB, b2);

    // output head: out = x2 @ Wout + bout
    gemm_out_wmma<<<ntiles, 96, 0, stream>>>(bufB, Wout, bout, out);
}